// SubwordPooling_20444044329685
// MI455X (gfx1250) — compile-verified
//
#include <hip/hip_runtime.h>
#include <hip/hip_bf16.h>

typedef __attribute__((ext_vector_type(2))) float v2f;
typedef __attribute__((ext_vector_type(8))) float v8f;

#define SP_B 32
#define SP_S 4096
#define SP_E 256
#define SP_W 2048
#define SP_WTILE 16                 // words per block tile
#define SP_NTILES (SP_W / SP_WTILE) // 128 tiles per batch row

// One block per (batch, 16-word tile). 256 threads = 8 waves; each wave owns a
// 32-wide slice of E as two 16x16 f32 WMMA accumulators.
// D(16w x 16e) += A(16w x 4k) * B(4k x 16e), A = one-hot(word)/count weights.
// ids sorted per row -> each tile's tokens are one contiguous run [start,end).
__global__ __launch_bounds__(256) void SubwordPooling_kernel(
    const float* __restrict__ emb,   // (B, S, E)
    const int*   __restrict__ ids,   // (B, S) sorted non-decreasing
    float*       __restrict__ out)   // (B, W, E)
{
    const int blk   = blockIdx.x;
    const int b     = blk >> 7;          // / SP_NTILES
    const int wtile = blk & (SP_NTILES - 1);
    const int w0    = wtile * SP_WTILE;

    const int*   idrow  = ids + (size_t)b * SP_S;
    const float* embrow = emb + (size_t)b * SP_S * SP_E;
    float*       outrow = out + (size_t)b * SP_W * SP_E;

    __shared__ int   bounds[SP_WTILE + 1];
    __shared__ float recip[SP_WTILE];

    const int tid = threadIdx.x;

    // lower_bound for word ids w0 .. w0+16 -> run boundaries (12 steps, S=4096)
    if (tid < SP_WTILE + 1) {
        const int target = w0 + tid;
        int lo = 0, hi = SP_S;
        while (lo < hi) {
            const int mid = (lo + hi) >> 1;
            if (idrow[mid] < target) lo = mid + 1; else hi = mid;
        }
        bounds[tid] = lo;
    }
    __syncthreads();
    if (tid < SP_WTILE) {
        const int c = bounds[tid + 1] - bounds[tid];
        recip[tid] = (c > 0) ? (1.0f / (float)c) : 0.0f;
    }
    __syncthreads();

    const int start = bounds[0];
    const int end   = bounds[SP_WTILE];

    const int wave   = tid >> 5;          // 0..7
    const int lane   = tid & 31;
    const int lane16 = lane & 15;
    const int khalf  = lane >> 4;         // lane half selects K={0,1} or K={2,3}
    const int kbase  = khalf << 1;
    const int col0   = wave * 32 + lane16;   // this lane's B/D column (tile 0)
    const float wrecip = recip[lane16];
    const int   myword = w0 + lane16;        // A-matrix row M = lane%16

    v8f acc0 = {};
    v8f acc1 = {};

    // Pointer-carried induction: one 64-bit bump per stream per chunk; the
    // 4 emb elements of a chunk live at fixed immediate offsets from bptr:
    //   +0 (K=kbase,col0)  +SP_E (K=kbase+1,col0)  +16 / +SP_E+16 (col1)
    const float* bptr = embrow + (size_t)(start + kbase) * SP_E + col0;
    const int*   iptr = idrow + start + kbase;

    // ---- guard-free main loop: whole K=4 chunks, no range checks ----
    int base = start;
#pragma unroll 2
    for (; base + 4 <= end; base += 4) {
        const int id0 = iptr[0];
        const int id1 = iptr[1];

        v2f a;
        a.x = (id0 == myword) ? wrecip : 0.0f;
        a.y = (id1 == myword) ? wrecip : 0.0f;

        v2f b0, b1;
        b0.x = __builtin_nontemporal_load(bptr);
        b0.y = __builtin_nontemporal_load(bptr + SP_E);
        b1.x = __builtin_nontemporal_load(bptr + 16);
        b1.y = __builtin_nontemporal_load(bptr + SP_E + 16);

        acc0 = __builtin_amdgcn_wmma_f32_16x16x4_f32(
            false, a, false, b0, (short)0, acc0, false, false);
        acc1 = __builtin_amdgcn_wmma_f32_16x16x4_f32(
            false, a, false, b1, (short)0, acc1, false, false);

        bptr += 4 * SP_E;
        iptr += 4;
    }

    // ---- single guarded tail chunk (1..3 tokens remain) ----
    if (base < end) {
        const int endm1 = end - 1;
        const int i0 = base + kbase;
        const int i1 = i0 + 1;
        const int c0 = (i0 < end) ? i0 : endm1;   // clamped, in-bounds
        const int c1 = (i1 < end) ? i1 : endm1;

        const int id0 = idrow[c0];
        const int id1 = idrow[c1];

        v2f a;
        a.x = ((i0 < end) & (id0 == myword)) ? wrecip : 0.0f;
        a.y = ((i1 < end) & (id1 == myword)) ? wrecip : 0.0f;

        const float* t0 = embrow + (size_t)c0 * SP_E + col0;
        const float* t1 = embrow + (size_t)c1 * SP_E + col0;
        v2f b0, b1;
        b0.x = __builtin_nontemporal_load(t0);
        b0.y = __builtin_nontemporal_load(t1);
        b1.x = __builtin_nontemporal_load(t0 + 16);
        b1.y = __builtin_nontemporal_load(t1 + 16);

        acc0 = __builtin_amdgcn_wmma_f32_16x16x4_f32(
            false, a, false, b0, (short)0, acc0, false, false);
        acc1 = __builtin_amdgcn_wmma_f32_16x16x4_f32(
            false, a, false, b1, (short)0, acc1, false, false);
    }

    // D layout (16x16 f32): VGPR j -> row M = j + 8*(lane/16), col N = lane%16
    float* optr = outrow + (size_t)(w0 + (khalf << 3)) * SP_E + col0;
#pragma unroll
    for (int j = 0; j < 8; ++j) {
        __builtin_nontemporal_store(acc0[j], optr + (size_t)j * SP_E);
        __builtin_nontemporal_store(acc1[j], optr + (size_t)j * SP_E + 16);
    }
}

extern "C" void kernel_launch(void* const* d_in, const int* in_sizes, int n_in,
                              void* d_out, int out_size, void* d_ws, size_t ws_size,
                              hipStream_t stream) {
    (void)in_sizes; (void)n_in; (void)out_size; (void)d_ws; (void)ws_size;
    const float* emb = (const float*)d_in[0];   // (B,S,E) f32
    const int*   ids = (const int*)d_in[1];     // (B,S) i32
    float*       out = (float*)d_out;           // (B,W,E) f32

    dim3 grid(SP_B * SP_NTILES);  // 4096 blocks
    dim3 block(256);              // 8 waves
    SubwordPooling_kernel<<<grid, block, 0, stream>>>(emb, ids, out);
}